// SimpleDecoder_2680059593232
// MI455X (gfx1250) — compile-verified
//
#include <hip/hip_runtime.h>

typedef float v2f __attribute__((ext_vector_type(2)));
typedef float v4f __attribute__((ext_vector_type(4)));
typedef float v8f __attribute__((ext_vector_type(8)));

#define BB 4
#define LL 4096
#define DD 1024
#define MM 512

// ---------------- kernel 1: p_at_bounds = clip(where(mask, bp[b, bidx], 1), 0.1) ----------
__global__ void k_pbounds(const float* __restrict__ bp, const int* __restrict__ bidx,
                          const unsigned char* __restrict__ mask, float* __restrict__ wp) {
  int i = blockIdx.x * blockDim.x + threadIdx.x;
  if (i >= BB * MM) return;
  int b = i >> 9;                       // MM = 512
  int bi = bidx[i];
  float p = bp[(b << 12) + bi];         // LL = 4096
  p = mask[i] ? p : 1.0f;
  wp[i] = fmaxf(p, 0.1f);
}

// ---------------- kernel 2: EMA scan over m; one thread per (b, d) ----------------------
__global__ void k_smooth(const float* __restrict__ ct, const float* __restrict__ wp,
                         float* __restrict__ sm) {
  __shared__ float sp[MM];
  const int b = blockIdx.x >> 2;                        // 16 blocks, 4 per batch
  const int d = ((blockIdx.x & 3) << 8) + threadIdx.x;  // 4*256 = 1024 d-lanes
  sp[threadIdx.x]       = wp[b * MM + threadIdx.x];
  sp[threadIdx.x + 256] = wp[b * MM + threadIdx.x + 256];
  __syncthreads();
  const float* cb = ct + ((size_t)b << 19) + d;         // 512*1024 = 2^19
  float*       sb = sm + ((size_t)b << 19) + d;
  float h = cb[0];
  sb[0] = h;
  for (int m = 1; m < MM; ++m) {
    float pv    = sp[m];
    float decay = fmaxf(1.0f - pv, 1e-7f);
    h = decay * h + pv * cb[(size_t)m * DD];
    sb[(size_t)m * DD] = h;
  }
}

// ---------------- kernel 3: plug_back_idx = clip(cumsum(bp>=0.5) - 1, 0) ----------------
__global__ __launch_bounds__(1024) void k_plugidx(const float* __restrict__ bp,
                                                  int* __restrict__ pidx) {
  __shared__ int ssum[1024];
  const int b = blockIdx.x;
  const int t = threadIdx.x;
  const float* p = bp + (size_t)b * LL;
  int m[4], s = 0;
#pragma unroll
  for (int j = 0; j < 4; ++j) {
    m[j] = (p[t * 4 + j] >= 0.5f) ? 1 : 0;
    s += m[j];
  }
  ssum[t] = s;
  __syncthreads();
  for (int off = 1; off < 1024; off <<= 1) {
    int v = 0;
    if (t >= off) v = ssum[t - off];
    __syncthreads();
    ssum[t] += v;
    __syncthreads();
  }
  int run = ssum[t] - s;  // exclusive prefix over threads
#pragma unroll
  for (int j = 0; j < 4; ++j) {
    run += m[j];
    int idx = run - 1;
    pidx[(size_t)b * LL + t * 4 + j] = idx < 0 ? 0 : idx;
  }
}

// ---------------- kernel 4: out = enc @ W^T + gather(smoothed, plugidx) -----------------
// Workgroup tile 128(M) x 64(N); 8 waves, each owns a 32x32 region (2x2 WMMA tiles).
// K staged in chunks of 16 through double-buffered LDS (one barrier per chunk).
// Padded stride 20 floats: 16B-aligned b128 stores, bank-conflict-free fragment reads.
#define TM 128
#define TN 64
#define LSTR 20
#define ABUF (TM * LSTR)
#define BBUF (TN * LSTR)

__global__ __launch_bounds__(256) void k_gemm(const float* __restrict__ enc,
                                              const float* __restrict__ W,
                                              const float* __restrict__ sm,
                                              const int* __restrict__ pidx,
                                              float* __restrict__ out) {
  __shared__ __align__(16) float lA[2 * ABUF];
  __shared__ __align__(16) float lB[2 * BBUF];

  const int t    = threadIdx.x;
  const int lane = t & 31;
  const int wave = t >> 5;
  const int wM   = (wave >> 1) * 32;   // 0,32,64,96
  const int wN   = (wave & 1) * 32;    // 0,32
  const int bM   = blockIdx.y * TM;    // row base into flattened (B*L)
  const int bN   = blockIdx.x * TN;    // col base into D

  const int lrow  = t >> 2;            // 0..63
  const int lkv   = (t & 3) * 4;       // 0,4,8,12 (floats within 16-wide K chunk)
  const int ml    = lane & 15;
  const int khalf = (lane >> 4) << 1;  // 0 or 2 (fragment K sub-offset per half-wave)

  v8f acc00 = {0,0,0,0,0,0,0,0};
  v8f acc01 = {0,0,0,0,0,0,0,0};
  v8f acc10 = {0,0,0,0,0,0,0,0};
  v8f acc11 = {0,0,0,0,0,0,0,0};

  const float* aptr0 = enc + (size_t)(bM + lrow) * DD + lkv;
  const float* aptr1 = enc + (size_t)(bM + 64 + lrow) * DD + lkv;
  const float* bptr  = W   + (size_t)(bN + lrow) * DD + lkv;   // B[k][n] = W[n*D + k]

  v4f ra0, ra1, rb0;

#define GLOAD(K0)                                      \
  do {                                                 \
    ra0 = *(const v4f*)(aptr0 + (K0));                 \
    ra1 = *(const v4f*)(aptr1 + (K0));                 \
    rb0 = *(const v4f*)(bptr  + (K0));                 \
  } while (0)

#define LSTORE(OFFA, OFFB)                                         \
  do {                                                             \
    *(v4f*)&lA[(OFFA) + lrow * LSTR + lkv]        = ra0;           \
    *(v4f*)&lA[(OFFA) + (64 + lrow) * LSTR + lkv] = ra1;           \
    *(v4f*)&lB[(OFFB) + lrow * LSTR + lkv]        = rb0;           \
  } while (0)

#define COMPUTE(OFFA, OFFB)                                                                  \
  {                                                                                          \
    _Pragma("unroll")                                                                        \
    for (int kk = 0; kk < 4; ++kk) {                                                         \
      const int kb  = kk * 4 + khalf;                                                        \
      const int ia0 = (OFFA) + (wM + ml) * LSTR + kb;                                        \
      const int ib0 = (OFFB) + (wN + ml) * LSTR + kb;                                        \
      v2f af0 = { lA[ia0],             lA[ia0 + 1]             };                            \
      v2f af1 = { lA[ia0 + 16 * LSTR], lA[ia0 + 16 * LSTR + 1] };                            \
      v2f bf0 = { lB[ib0],             lB[ib0 + 1]             };                            \
      v2f bf1 = { lB[ib0 + 16 * LSTR], lB[ib0 + 16 * LSTR + 1] };                            \
      acc00 = __builtin_amdgcn_wmma_f32_16x16x4_f32(false, af0, false, bf0, (short)0, acc00, false, false); \
      acc01 = __builtin_amdgcn_wmma_f32_16x16x4_f32(false, af0, false, bf1, (short)0, acc01, false, false); \
      acc10 = __builtin_amdgcn_wmma_f32_16x16x4_f32(false, af1, false, bf0, (short)0, acc10, false, false); \
      acc11 = __builtin_amdgcn_wmma_f32_16x16x4_f32(false, af1, false, bf1, (short)0, acc11, false, false); \
    }                                                                                        \
  }

  // Prologue: chunk 0 into buffer 0.
  GLOAD(0);
  LSTORE(0, 0);
  __syncthreads();

  // 64 K-chunks of 16; 2 chunks (even->buf0, odd->buf1) per loop trip; 1 barrier per chunk.
  for (int kc = 0; kc < 64; kc += 2) {
    GLOAD((kc + 1) * 16);                       // in flight under the WMMA block
    if (kc + 3 < 64) {                          // pull chunk kc+3 toward L2
      __builtin_prefetch(aptr0 + (kc + 3) * 16, 0, 0);
      __builtin_prefetch(aptr1 + (kc + 3) * 16, 0, 0);
      __builtin_prefetch(bptr  + (kc + 3) * 16, 0, 0);
    }
    COMPUTE(0, 0);                              // read buf0 (chunk kc)
    LSTORE(ABUF, BBUF);                         // fill buf1 (chunk kc+1)
    __syncthreads();

    if (kc + 2 < 64) GLOAD((kc + 2) * 16);
    COMPUTE(ABUF, BBUF);                        // read buf1 (chunk kc+1)
    if (kc + 2 < 64) LSTORE(0, 0);              // refill buf0 (chunk kc+2)
    __syncthreads();
  }

  // Epilogue: C/D layout row = r + 8*(lane>>4), col = lane&15. Fuse plugback gather.
  const int rh = (lane >> 4) << 3;  // 0 or 8

#define EPILOGUE(ACC, MT, NT)                                            \
  {                                                                      \
    const int gm = bM + wM + (MT) * 16 + rh;                             \
    const int gn = bN + wN + (NT) * 16 + ml;                             \
    _Pragma("unroll")                                                    \
    for (int r = 0; r < 8; ++r) {                                        \
      const int row = gm + r;            /* row = b*L + l */             \
      const int bb2 = row >> 12;         /* L = 4096 */                  \
      const int pi  = pidx[row];                                         \
      const float pb = sm[((size_t)bb2 << 19) + (size_t)pi * DD + gn];   \
      out[(size_t)row * DD + gn] = ACC[r] + pb;                          \
    }                                                                    \
  }

  EPILOGUE(acc00, 0, 0);
  EPILOGUE(acc01, 0, 1);
  EPILOGUE(acc10, 1, 0);
  EPILOGUE(acc11, 1, 1);

#undef GLOAD
#undef LSTORE
#undef COMPUTE
#undef EPILOGUE
}

extern "C" void kernel_launch(void* const* d_in, const int* in_sizes, int n_in,
                              void* d_out, int out_size, void* d_ws, size_t ws_size,
                              hipStream_t stream) {
  const float*         concept_tokens = (const float*)d_in[0];          // (B, M, D)
  const float*         encoder_out    = (const float*)d_in[1];          // (B, L, D)
  const float*         boundary_probs = (const float*)d_in[2];          // (B, L)
  const int*           boundary_idx   = (const int*)d_in[3];            // (B, M)
  const unsigned char* concept_mask   = (const unsigned char*)d_in[4];  // (B, M) bool
  const float*         Wmat           = (const float*)d_in[5];          // (D, D)
  float* out = (float*)d_out;

  char*  ws   = (char*)d_ws;
  float* wp   = (float*)ws;                    //  2048 floats (8 KB)
  int*   widx = (int*)(ws + 8192);             // 16384 ints  (64 KB)
  float* wsm  = (float*)(ws + 8192 + 65536);   // 2^21 floats (8 MB) smoothed

  k_pbounds<<<dim3((BB * MM + 255) / 256), dim3(256), 0, stream>>>(
      boundary_probs, boundary_idx, concept_mask, wp);
  k_smooth<<<dim3(BB * 4), dim3(256), 0, stream>>>(concept_tokens, wp, wsm);
  k_plugidx<<<dim3(BB), dim3(1024), 0, stream>>>(boundary_probs, widx);

  dim3 grid(DD / TN, (BB * LL) / TM);  // (16, 128)
  k_gemm<<<grid, dim3(256), 0, stream>>>(encoder_out, Wmat, wsm, widx, out);
}